// cWCT_8744553414811
// MI455X (gfx1250) — compile-verified
//
#include <hip/hip_runtime.h>

typedef __attribute__((ext_vector_type(2))) float v2f;
typedef __attribute__((ext_vector_type(8))) float v8f;

#define B_ 8
#define N_ 256
#define L_ 16384
#define EPS_ 2e-5f

// ---------------- Kernel 1: per-row means (content & style) ----------------
__global__ void k_means(const float* __restrict__ cont, const float* __restrict__ styl,
                        float* __restrict__ mean_c, float* __restrict__ mean_s) {
  int id = blockIdx.x;                      // 0 .. 2*B*N-1
  int which = (id >= B_ * N_);
  int row = which ? id - B_ * N_ : id;
  const float* x = (which ? styl : cont) + (size_t)row * L_;
  float s = 0.f;
  for (int i = threadIdx.x; i < L_; i += 256) s += x[i];
  __shared__ float red[256];
  red[threadIdx.x] = s;
  __syncthreads();
  for (int off = 128; off > 0; off >>= 1) {
    if (threadIdx.x < off) red[threadIdx.x] += red[threadIdx.x + off];
    __syncthreads();
  }
  if (threadIdx.x == 0) (which ? mean_s : mean_c)[row] = red[0] * (1.0f / L_);
}

// ---------------- Kernel 2: WMMA Gram -> covariance -------------------------
// Cov = (X X^T - L*mu mu^T)/(L-1) + eps*I.  64x64 macro-tile per workgroup,
// 4 waves x 32x32 block (2x2 accumulators, 1 fragment load per WMMA).
// K-chunks are double-buffered in LDS and filled with
// GLOBAL_LOAD_ASYNC_TO_LDS_B128 (ASYNCcnt) so the fill overlaps the WMMAs.
#define KC 32
#define LDP 36                  // floats: 16B-aligned, 16 distinct banks/16 rows
#define BUF (64 * LDP)          // floats per buffer per matrix side
__global__ __launch_bounds__(128) void k_gram(
    const float* __restrict__ cont, const float* __restrict__ styl,
    const float* __restrict__ mean_c, const float* __restrict__ mean_s,
    float* __restrict__ cov_c, float* __restrict__ cov_s) {
  int id = blockIdx.x;                       // 2 * B * 16 macro-tiles (64x64)
  int which = (id >= B_ * 16);
  int r = which ? id - B_ * 16 : id;
  int b = r >> 4;
  int tile = r & 15;
  int tm = tile >> 2, tn = tile & 3;
  const float* X = (which ? styl : cont) + (size_t)b * N_ * L_;
  const float* mean = (which ? mean_s : mean_c) + b * N_;
  float* cov = (which ? cov_s : cov_c) + (size_t)b * N_ * N_;

  __shared__ alignas(16) float Al[2 * BUF];
  __shared__ alignas(16) float Bl[2 * BUF];

  const int tid = threadIdx.x, lane = tid & 31, wave = tid >> 5;
  const int qm = (wave >> 1) * 32, qn = (wave & 1) * 32;   // 32x32 block / wave
  const int rowA0 = tm * 64, rowB0 = tn * 64;
  const int a0r = qm + (lane & 15), a1r = a0r + 16;
  const int b0r = qn + (lane & 15), b1r = b0r + 16;
  const int koff = (lane >> 4) * 2;          // lanes 16..31 hold K+2,K+3

  // Per-thread async-copy slots: 4 float4's for A, 4 for B, per chunk.
  // e = tid + 128*j ; row = e/8 ; c4 = (e%8)*4 floats.
  unsigned int gA[4], gB[4], lA[4], lB[4];
#pragma unroll
  for (int j = 0; j < 4; ++j) {
    int e = tid + 128 * j;
    unsigned int row = (unsigned)(e >> 3);
    unsigned int c4 = (unsigned)((e & 7) << 2);
    gA[j] = (((unsigned)(rowA0) + row) * (unsigned)L_ + c4) * 4u;  // byte off
    gB[j] = (((unsigned)(rowB0) + row) * (unsigned)L_ + c4) * 4u;
    unsigned int loff = (row * LDP + c4) * 4u;                      // byte off
    lA[j] = (unsigned int)(uintptr_t)(&Al[0]) + loff;  // low32 = LDS offset
    lB[j] = (unsigned int)(uintptr_t)(&Bl[0]) + loff;
  }

  const int nch = L_ / KC;
  // prologue: fill buffer 0 with chunk 0
  {
    const char* base = (const char*)X;
#pragma unroll
    for (int j = 0; j < 4; ++j) {
      asm volatile("global_load_async_to_lds_b128 %0, %1, %2"
                   :: "v"(lA[j]), "v"(gA[j]), "s"(base) : "memory");
      asm volatile("global_load_async_to_lds_b128 %0, %1, %2"
                   :: "v"(lB[j]), "v"(gB[j]), "s"(base) : "memory");
    }
  }

  v8f acc00 = {}, acc01 = {}, acc10 = {}, acc11 = {};
  for (int c = 0; c < nch; ++c) {
    const int cur = (c & 1) * BUF;
    if (c + 1 < nch) {                       // prefetch next chunk, other buffer
      const char* base = (const char*)(X + (size_t)(c + 1) * KC);
      const unsigned int add = (unsigned)(((c + 1) & 1) * BUF * 4);
#pragma unroll
      for (int j = 0; j < 4; ++j) {
        asm volatile("global_load_async_to_lds_b128 %0, %1, %2"
                     :: "v"(lA[j] + add), "v"(gA[j]), "s"(base) : "memory");
        asm volatile("global_load_async_to_lds_b128 %0, %1, %2"
                     :: "v"(lB[j] + add), "v"(gB[j]), "s"(base) : "memory");
      }
      // async completes in-order: <=8 outstanding means current chunk landed
      asm volatile("s_wait_asynccnt 0x8" ::: "memory");
    } else {
      asm volatile("s_wait_asynccnt 0x0" ::: "memory");
    }
    __syncthreads();
#pragma unroll
    for (int k = 0; k < KC; k += 4) {
      v2f a0 = *(const v2f*)&Al[cur + a0r * LDP + k + koff];
      v2f a1 = *(const v2f*)&Al[cur + a1r * LDP + k + koff];
      v2f b0 = *(const v2f*)&Bl[cur + b0r * LDP + k + koff];
      v2f b1 = *(const v2f*)&Bl[cur + b1r * LDP + k + koff];
      acc00 = __builtin_amdgcn_wmma_f32_16x16x4_f32(false, a0, false, b0, (short)0, acc00, false, false);
      acc01 = __builtin_amdgcn_wmma_f32_16x16x4_f32(false, a0, false, b1, (short)0, acc01, false, false);
      acc10 = __builtin_amdgcn_wmma_f32_16x16x4_f32(false, a1, false, b0, (short)0, acc10, false, false);
      acc11 = __builtin_amdgcn_wmma_f32_16x16x4_f32(false, a1, false, b1, (short)0, acc11, false, false);
    }
    __syncthreads();   // nobody refills this buffer until all waves are done
  }

  // epilogue: rank-1 mean correction + eps jitter
  const float scale = 1.0f / (float)(L_ - 1);
  const int hi8 = (lane >> 4) << 3;
  int nc0 = tn * 64 + qn + (lane & 15);
  int nc1 = nc0 + 16;
  float mn0 = mean[nc0], mn1 = mean[nc1];
#pragma unroll
  for (int rr = 0; rr < 8; ++rr) {
    int mr0 = tm * 64 + qm + rr + hi8;
    int mr1 = mr0 + 16;
    float mm0 = mean[mr0], mm1 = mean[mr1];
    float v00 = (acc00[rr] - (float)L_ * mm0 * mn0) * scale;
    float v01 = (acc01[rr] - (float)L_ * mm0 * mn1) * scale;
    float v10 = (acc10[rr] - (float)L_ * mm1 * mn0) * scale;
    float v11 = (acc11[rr] - (float)L_ * mm1 * mn1) * scale;
    if (mr0 == nc0) v00 += EPS_;
    if (mr0 == nc1) v01 += EPS_;
    if (mr1 == nc0) v10 += EPS_;
    if (mr1 == nc1) v11 += EPS_;
    cov[mr0 * N_ + nc0] = v00;
    cov[mr0 * N_ + nc1] = v01;
    cov[mr1 * N_ + nc0] = v10;
    cov[mr1 * N_ + nc1] = v11;
  }
}

// ---------------- Kernel 3: in-place Cholesky (one WG per 256x256 matrix) ---
__global__ void k_chol(float* __restrict__ cov_c, float* __restrict__ cov_s) {
  int id = blockIdx.x;  // 2*B
  float* A = (id >= B_) ? cov_s + (size_t)(id - B_) * N_ * N_
                        : cov_c + (size_t)id * N_ * N_;
  __shared__ float sdiag;
  int tid = threadIdx.x;
  for (int k = 0; k < N_; ++k) {
    if (tid == 0) sdiag = sqrtf(A[k * N_ + k]);
    __syncthreads();
    float inv = 1.0f / sdiag;
    for (int i = k + 1 + tid; i < N_; i += 256) A[i * N_ + k] *= inv;
    if (tid == 0) A[k * N_ + k] = sdiag;
    __syncthreads();
    for (int i = k + 1 + tid; i < N_; i += 256) {
      float lik = A[i * N_ + k];
      for (int j = k + 1; j <= i; ++j) A[i * N_ + j] -= lik * A[j * N_ + k];
    }
    __syncthreads();
  }
  for (int e = tid; e < N_ * N_; e += 256) {   // zero strict upper triangle
    int i = e >> 8, j = e & 255;
    if (j > i) A[e] = 0.f;
  }
}

// ---------------- Kernel 4: M = Ls * inv(Lc) (solve M*Lc = Ls), bias --------
__global__ void k_minv(const float* __restrict__ cov_c, const float* __restrict__ cov_s,
                       const float* __restrict__ mean_c, const float* __restrict__ mean_s,
                       float* __restrict__ Mw, float* __restrict__ bias) {
  int b = blockIdx.x;
  int r = threadIdx.x;
  const float* Lc = cov_c + (size_t)b * N_ * N_;
  const float* Ls = cov_s + (size_t)b * N_ * N_;
  float* x = Mw + (size_t)b * N_ * N_ + (size_t)r * N_;
  for (int j = 0; j < N_; ++j) x[j] = (j <= r) ? Ls[r * N_ + j] : 0.f;
  for (int j = r; j >= 0; --j) {           // backward substitution, M lower-tri
    float s = x[j];
    for (int k = j + 1; k <= r; ++k) s -= x[k] * Lc[k * N_ + j];
    x[j] = s / Lc[j * N_ + j];
  }
  const float* cm = mean_c + b * N_;
  float dotv = 0.f;
  for (int j = 0; j <= r; ++j) dotv += x[j] * cm[j];
  bias[b * N_ + r] = (mean_s + b * N_)[r] - dotv;
}

// ---------------- Kernel 5: out = M @ X + bias (WMMA, K = 256) --------------
#define MLDP 260               // padded LDS stride for 32 staged M-rows
__global__ __launch_bounds__(128) void k_color(
    const float* __restrict__ cont, const float* __restrict__ Mw,
    const float* __restrict__ bias, float* __restrict__ out) {
  int id = blockIdx.x;                        // B * (N/32) * (L/128)
  const int per_b = (N_ / 32) * (L_ / 128);
  int b = id / per_b;
  int rr = id % per_b;
  int mt = rr / (L_ / 128);
  int lt = rr % (L_ / 128);
  const float* Xb = cont + (size_t)b * N_ * L_;
  const float* Mb = Mw + (size_t)b * N_ * N_ + (size_t)mt * 32 * N_;
  float* outb = out + (size_t)b * N_ * L_;

  __shared__ alignas(16) float Ml[32 * MLDP];
  int tid = threadIdx.x, lane = tid & 31, wave = tid >> 5;
  for (int e = tid; e < 32 * (N_ / 4); e += 128) {   // float4 stage of M rows
    int row = e >> 6, c4 = (e & 63) << 2;
    *(float4*)&Ml[row * MLDP + c4] = *(const float4*)&Mb[row * N_ + c4];
  }
  __syncthreads();

  const int l0 = lt * 128 + wave * 32 + (lane & 15);
  const int l1 = l0 + 16;
  const int koff = (lane >> 4) * 2;
  const int a0r = lane & 15, a1r = a0r + 16;
  v8f acc00 = {}, acc01 = {}, acc10 = {}, acc11 = {};
#pragma unroll 4
  for (int k = 0; k < N_; k += 4) {
    v2f a0 = *(const v2f*)&Ml[a0r * MLDP + k + koff];
    v2f a1 = *(const v2f*)&Ml[a1r * MLDP + k + koff];
    v2f b0, b1;
    const float* row0 = Xb + (size_t)(k + koff) * L_;
    const float* row1 = Xb + (size_t)(k + koff + 1) * L_;
    b0.x = row0[l0];  b0.y = row1[l0];      // coalesced across lanes 0..15
    b1.x = row0[l1];  b1.y = row1[l1];
    acc00 = __builtin_amdgcn_wmma_f32_16x16x4_f32(false, a0, false, b0, (short)0, acc00, false, false);
    acc01 = __builtin_amdgcn_wmma_f32_16x16x4_f32(false, a0, false, b1, (short)0, acc01, false, false);
    acc10 = __builtin_amdgcn_wmma_f32_16x16x4_f32(false, a1, false, b0, (short)0, acc10, false, false);
    acc11 = __builtin_amdgcn_wmma_f32_16x16x4_f32(false, a1, false, b1, (short)0, acc11, false, false);
  }
  const int hi8 = (lane >> 4) << 3;
#pragma unroll
  for (int v = 0; v < 8; ++v) {
    int m0 = mt * 32 + v + hi8;
    int m1 = m0 + 16;
    float bi0 = bias[b * N_ + m0], bi1 = bias[b * N_ + m1];
    outb[(size_t)m0 * L_ + l0] = acc00[v] + bi0;
    outb[(size_t)m0 * L_ + l1] = acc01[v] + bi0;
    outb[(size_t)m1 * L_ + l0] = acc10[v] + bi1;
    outb[(size_t)m1 * L_ + l1] = acc11[v] + bi1;
  }
}

// ---------------- Host launcher --------------------------------------------
extern "C" void kernel_launch(void* const* d_in, const int* in_sizes, int n_in,
                              void* d_out, int out_size, void* d_ws, size_t ws_size,
                              hipStream_t stream) {
  const float* cont = (const float*)d_in[0];
  const float* styl = (const float*)d_in[1];
  float* out = (float*)d_out;
  float* ws = (float*)d_ws;

  float* mean_c = ws;                                   // B*N
  float* mean_s = mean_c + B_ * N_;                     // B*N
  float* cov_c  = mean_s + B_ * N_;                     // B*N*N
  float* cov_s  = cov_c + (size_t)B_ * N_ * N_;         // B*N*N
  float* Mw     = cov_s + (size_t)B_ * N_ * N_;         // B*N*N
  float* bias   = Mw + (size_t)B_ * N_ * N_;            // B*N

  k_means<<<2 * B_ * N_, 256, 0, stream>>>(cont, styl, mean_c, mean_s);
  k_gram<<<2 * B_ * 16, 128, 0, stream>>>(cont, styl, mean_c, mean_s, cov_c, cov_s);
  k_chol<<<2 * B_, 256, 0, stream>>>(cov_c, cov_s);
  k_minv<<<B_, 256, 0, stream>>>(cov_c, cov_s, mean_c, mean_s, Mw, bias);
  k_color<<<B_ * (N_ / 32) * (L_ / 128), 128, 0, stream>>>(cont, Mw, bias, out);
}